// NonMaximumSuppression_6597069767423
// MI455X (gfx1250) — compile-verified
//
#include <hip/hip_runtime.h>
#include <cstdint>
#include <cstddef>

// ---------------- problem constants (from reference) ----------------
#define NBOX   100000
#define NCLS   20
#define ITERS  300
#define SCORE_THR 0.05f
#define NMS_THR   0.5f

// ---------------- kernel geometry ----------------
#define NWG    4                       // workgroups cooperating per class
#define NPART  (NBOX / NWG)            // 25000 boxes per workgroup
#define BLK    1024                    // 32 wave32 waves per workgroup
#define KPT    ((NPART + BLK - 1) / BLK) // 25 score slots per thread (in VGPRs)
#define TILE   4                       // elements per async double-buffer tile
#define STAGE_BYTES (2 * TILE * BLK * 16)  // 128 KB LDS box staging

typedef int v4i __attribute__((vector_size(16)));
typedef __attribute__((address_space(1))) v4i* gptr_v4i;   // global (device) pointer
typedef __attribute__((address_space(3))) v4i* lptr_v4i;   // LDS pointer

// ---------------- CDNA5 async global->LDS path ----------------
#if __has_builtin(__builtin_amdgcn_global_load_async_to_lds_b128)
#define HAVE_ASYNC 1
#else
#define HAVE_ASYNC 0
#endif

#if __has_builtin(__builtin_amdgcn_s_wait_asynccnt)
#define WAIT_ASYNC(n) __builtin_amdgcn_s_wait_asynccnt(n)
#elif HAVE_ASYNC
#define WAIT_ASYNC(n) asm volatile("s_wait_asynccnt %0" :: "i"(n) : "memory")
#else
#define WAIT_ASYNC(n) do {} while (0)
#endif

__device__ __forceinline__ void async_copy16(const float* g, void* l) {
#if HAVE_ASYNC
  __builtin_amdgcn_global_load_async_to_lds_b128((gptr_v4i)g, (lptr_v4i)l, 0, 0);
#else
  *(float4*)l = *(const float4*)g;   // synchronous fallback
#endif
}

// monotonic argmax key: larger score wins; tie -> smaller index (matches jnp.argmax)
__device__ __forceinline__ unsigned long long mk_key(float v, unsigned e) {
  unsigned u = __float_as_uint(v);
  u ^= (unsigned)(((int)u) >> 31) | 0x80000000u;   // branchless order-preserving map
  return ((unsigned long long)u << 32) | (unsigned long long)(unsigned)(~e);
}

// issue TILE async b128 loads of this tile's boxes into its LDS buffer.
// Unconditional + clamped address => EXEC all-ones => uniform ASYNCcnt per wave.
template<int BASE>
__device__ __forceinline__ void prefetch_tile(const float* boxes, int pbase, int nend,
                                              int t, char* stage) {
  constexpr int BUF = (BASE / TILE) & 1;
#pragma unroll
  for (int j = 0; j < TILE; ++j) {
    if (BASE + j < KPT) {
      int e = pbase + (BASE + j) * BLK + t;
      if (e >= nend) e = pbase;                          // clamp: always in-bounds, always issued
      async_copy16(boxes + (size_t)e * 4,
                   stage + (((BUF * TILE + j) * BLK + t) << 4));
    }
  }
}

// fused suppress + argmax over one tile, with prefetch of the next tile in flight
template<int BASE>
__device__ __forceinline__ void run_tile(float* s, float& bv, int& bi,
                                         const float* boxes, int pbase, int nend, int t,
                                         char* stage, float wx1, float wy1, float wx2,
                                         float wy2, float warea) {
  constexpr int NXT = BASE + TILE;
  if constexpr (NXT < KPT) prefetch_tile<NXT>(boxes, pbase, nend, t, stage);
  constexpr int REMAIN = KPT - NXT;
  constexpr int NISSUE = (REMAIN > 0) ? (REMAIN < TILE ? REMAIN : TILE) : 0;
  asm volatile("" ::: "memory");
  WAIT_ASYNC(NISSUE);                 // everything older than next tile's issues is complete
  asm volatile("" ::: "memory");
  constexpr int BUF = (BASE / TILE) & 1;
#pragma unroll
  for (int j = 0; j < TILE; ++j) {
    if (BASE + j < KPT) {
      const int k = BASE + j;
      const float4 b = *(const float4*)(stage + (((BUF * TILE + j) * BLK + t) << 4));
      const int e = pbase + k * BLK + t;
      const float ix1 = fmaxf(wx1, b.x);
      const float iy1 = fmaxf(wy1, b.y);
      const float ix2 = fminf(wx2, b.z);
      const float iy2 = fminf(wy2, b.w);
      const float inter = fmaxf(ix2 - ix1, 0.f) * fmaxf(iy2 - iy1, 0.f);
      if (inter > 0.f) {                                  // rare path: boxes overlap winner
        const float areab = fmaxf(b.z - b.x, 0.f) * fmaxf(b.w - b.y, 0.f);
        const float uni = warea + areab - inter;
        const float iou = (uni > 0.f) ? (inter / uni) : 0.f;  // IEEE div: bit-exact vs reference
        if (iou > NMS_THR) s[k] = -__builtin_inff();
      }
      if (s[k] > bv) { bv = s[k]; bi = e; }   // strict '>' keeps smallest index on ties
    }
  }
}

template<int BASE>
__device__ __forceinline__ void run_all_tiles(float* s, float& bv, int& bi,
                                              const float* boxes, int pbase, int nend, int t,
                                              char* stage, float wx1, float wy1, float wx2,
                                              float wy2, float warea) {
  run_tile<BASE>(s, bv, bi, boxes, pbase, nend, t, stage, wx1, wy1, wx2, wy2, warea);
  if constexpr (BASE + TILE < KPT)
    run_all_tiles<BASE + TILE>(s, bv, bi, boxes, pbase, nend, t, stage, wx1, wy1, wx2, wy2, warea);
}

// zero d_out and the per-class barrier counters every call (graph-replay safe)
__global__ void nms_init_kernel(float* __restrict__ out, unsigned* __restrict__ counters,
                                int out_n) {
  const int i = blockIdx.x * blockDim.x + threadIdx.x;
  const int stride = gridDim.x * blockDim.x;
  for (int j = i; j < out_n; j += stride) out[j] = 0.f;
  if (i < NCLS) counters[i] = 0u;
}

__global__ __launch_bounds__(BLK, 1) void nms_persistent(
    const float* __restrict__ boxes, const float* __restrict__ cls, float* __restrict__ out,
    unsigned long long* __restrict__ partials, unsigned* __restrict__ counters) {
  __shared__ unsigned long long red[BLK / 32];
  __shared__ unsigned long long bc;
  __shared__ char stage[STAGE_BYTES];

  const int t = threadIdx.x;
  const int c = blockIdx.x / NWG;     // class
  const int p = blockIdx.x % NWG;     // partition within class
  const int pbase = p * NPART;
  const int nend = pbase + NPART;
  const int lane = t & 31;
  const int wid = t >> 5;

  // ---- load + threshold-filter this thread's scores into VGPRs ----
  float s[KPT];
  float bv = -__builtin_inff();
  int bi = 0;
#pragma unroll
  for (int k = 0; k < KPT; ++k) {
    const int e = pbase + k * BLK + t;
    float v = -__builtin_inff();
    if (e < nend) {
      const float sc = cls[(size_t)e * NCLS + c];
      if (sc > SCORE_THR) v = sc;
    }
    s[k] = v;
    if (v > bv) { bv = v; bi = e; }
  }

  unsigned long long* part = partials + (size_t)c * 2 * NWG;  // [parity][NWG]
  unsigned* cnt = counters + c;

  for (int ph = 1; ph <= ITERS; ++ph) {
    // start streaming tile 0 boxes while we do the argmax barrier
    prefetch_tile<0>(boxes, pbase, nend, t, stage);

    // ---- wave32 + block argmax reduction of u64 keys ----
    unsigned long long kk = mk_key(bv, (unsigned)bi);
#pragma unroll
    for (int off = 16; off > 0; off >>= 1) {
      const unsigned long long o = __shfl_xor(kk, off, 32);
      kk = o > kk ? o : kk;
    }
    if (lane == 0) red[wid] = kk;
    __syncthreads();
    if (wid == 0) {
      kk = red[lane];
#pragma unroll
      for (int off = 16; off > 0; off >>= 1) {
        const unsigned long long o = __shfl_xor(kk, off, 32);
        kk = o > kk ? o : kk;
      }
      if (lane == 0) {
        // ---- cross-workgroup (per-class) argmax via monotonic counter barrier ----
        const int par = ph & 1;
        __hip_atomic_store(&part[par * NWG + p], kk, __ATOMIC_RELAXED, __HIP_MEMORY_SCOPE_AGENT);
        __hip_atomic_fetch_add(cnt, 1u, __ATOMIC_ACQ_REL, __HIP_MEMORY_SCOPE_AGENT);
        const unsigned target = (unsigned)(NWG * ph);
        while (__hip_atomic_load(cnt, __ATOMIC_ACQUIRE, __HIP_MEMORY_SCOPE_AGENT) < target)
          __builtin_amdgcn_s_sleep(1);
        unsigned long long w = 0ull;
        for (int q = 0; q < NWG; ++q) {
          const unsigned long long v =
              __hip_atomic_load(&part[par * NWG + q], __ATOMIC_RELAXED, __HIP_MEMORY_SCOPE_AGENT);
          w = v > w ? v : w;
        }
        bc = w;
      }
    }
    __syncthreads();
    const unsigned long long wkey = bc;

    const unsigned hi = (unsigned)(wkey >> 32);
    if (!(hi & 0x80000000u)) break;            // no candidates left in this class
    const unsigned idx = ~(unsigned)wkey;      // winning box index
    const float wv = __uint_as_float(hi & 0x7FFFFFFFu);  // its (original) score
    if (t == 0 && idx >= (unsigned)pbase && idx < (unsigned)nend)
      out[(size_t)idx * NCLS + c] = wv;        // scatter into zero-initialized output

    const int sidx = __builtin_amdgcn_readfirstlane((int)idx);
    const float4 wb = *(const float4*)(boxes + (size_t)sidx * 4);
    const float warea = fmaxf(wb.z - wb.x, 0.f) * fmaxf(wb.w - wb.y, 0.f);

    // ---- fused: suppress IoU>thr (winner kills itself via IoU==1) + next argmax ----
    bv = -__builtin_inff();
    bi = 0;
    run_all_tiles<0>(s, bv, bi, boxes, pbase, nend, t, stage,
                     wb.x, wb.y, wb.z, wb.w, warea);
  }
}

extern "C" void kernel_launch(void* const* d_in, const int* in_sizes, int n_in,
                              void* d_out, int out_size, void* d_ws, size_t ws_size,
                              hipStream_t stream) {
  (void)in_sizes; (void)n_in; (void)ws_size;
  const float* boxes = (const float*)d_in[0];     // [1, N, 4] f32
  const float* cls   = (const float*)d_in[1];     // [1, N, C] f32
  float* out = (float*)d_out;                     // [1, N, C] f32

  // workspace layout: [NCLS][2 parity][NWG] u64 partials, then NCLS u32 counters
  unsigned long long* partials = (unsigned long long*)d_ws;
  unsigned* counters = (unsigned*)((char*)d_ws + (size_t)NCLS * 2 * NWG * sizeof(unsigned long long));

  nms_init_kernel<<<1024, 256, 0, stream>>>(out, counters, out_size);
  nms_persistent<<<NCLS * NWG, BLK, 0, stream>>>(boxes, cls, out, partials, counters);
}